// Detect_53017076302285
// MI455X (gfx1250) — compile-verified
//
#include <hip/hip_runtime.h>
#include <hip/hip_bf16.h>
#include <stdint.h>

#define NT 256
#define B_IMG 64
#define P_PRI 24564
#define C_CLS 21
#define TOPK 200
#define CONF_TH 0.1f
#define NMS_TH 0.45f
#define VAR0 0.1f
#define VAR1 0.2f
#define CHUNK 4096
#define PPAD 24576

// --- gfx1250 feature probes ---
#if defined(__has_builtin)
# if __has_builtin(__builtin_amdgcn_tensor_load_to_lds) && __has_builtin(__builtin_amdgcn_s_wait_tensorcnt)
#  define HAVE_TDM 1
# endif
# if __has_builtin(__builtin_amdgcn_global_load_async_to_lds_b32) && __has_builtin(__builtin_amdgcn_s_wait_asynccnt)
#  define HAVE_ASYNC_LDS 1
# endif
#endif
#ifndef HAVE_TDM
# define HAVE_TDM 0
# warning "gfx1250 TDM builtins not found"
#endif
#ifndef HAVE_ASYNC_LDS
# define HAVE_ASYNC_LDS 0
# warning "gfx1250 async-to-LDS builtins not found"
#endif

typedef __attribute__((address_space(1))) int g_int;   // global (device) addrspace
typedef __attribute__((address_space(3))) int l_int;   // LDS (shared) addrspace
typedef unsigned int u32x4 __attribute__((ext_vector_type(4)));
typedef int i32x4 __attribute__((ext_vector_type(4)));
typedef int i32x8 __attribute__((ext_vector_type(8)));

__device__ __forceinline__ unsigned mono(float f) {
  unsigned u = __float_as_uint(f);
  return u ^ ((u & 0x80000000u) ? 0xFFFFFFFFu : 0x80000000u);
}
__device__ __forceinline__ float unmono(unsigned k) {
  unsigned u = (k & 0x80000000u) ? (k ^ 0x80000000u) : ~k;
  return __uint_as_float(u);
}

// ---------------- kernel 0: zero output + per-image class masks -------------
__global__ void k_init(float* __restrict__ out, int n, unsigned* __restrict__ mask) {
  int i = blockIdx.x * blockDim.x + threadIdx.x;
  if (i < n) out[i] = 0.0f;
  if (i < B_IMG) mask[i] = 0u;
}

// ---------------- kernel 1: bandwidth sweep of conf -> class bitmask --------
__global__ __launch_bounds__(NT) void k_classmask(const float* __restrict__ conf,
                                                  unsigned* __restrict__ mask) {
  const int b = blockIdx.y;
  const int p = blockIdx.x * NT + threadIdx.x;
  __shared__ unsigned sm;
  if (threadIdx.x == 0) sm = 0u;
  __syncthreads();

  unsigned m = 0u;
  if (p < P_PRI) {
    const float* row = conf + ((size_t)b * P_PRI + p) * C_CLS;
    __builtin_prefetch(row, 0, 3);  // global_prefetch_b8 on gfx1250
#pragma unroll
    for (int c = 1; c < C_CLS; ++c) m |= (row[c] > CONF_TH) ? (1u << c) : 0u;
  }
  // wave32 OR-reduction (warpSize == 32 on CDNA5)
  for (int off = 16; off > 0; off >>= 1) m |= __shfl_xor(m, off, 32);
  if ((threadIdx.x & 31) == 0 && m) atomicOr(&sm, m);
  __syncthreads();
  if (threadIdx.x == 0 && sm) atomicOr(&mask[b], sm);
}

// ---------------- kernel 2: per-image top-K select + NMS + write ------------
struct SelSmem {
  unsigned long long cand[256];  // packed (key<<32)|(~idx) candidates
  unsigned keys[PPAD];           // monotone score keys (96 KB, needs CDNA5 320KB LDS)
  float    stage[CHUNK];         // async-landing pad for raw scores
  unsigned tie[512];             // indices tied at the threshold key
  unsigned hist[256];            // radix histogram
  float bx0[TOPK], by0[TOPK], bx1[TOPK], by1[TOPK], ar[TOPK], sc[TOPK];
  int  keepf[TOPK];
  int  validf[TOPK];
  unsigned nGT, nEQ, sPrefix, sNeed;
  int supflag;
};

__global__ __launch_bounds__(NT) void k_select(const float* __restrict__ loc,
                                               const float* __restrict__ conf,
                                               const float* __restrict__ prior,
                                               const unsigned* __restrict__ mask,
                                               float* __restrict__ out) {
  extern __shared__ unsigned char smem_raw[];
  SelSmem& S = *reinterpret_cast<SelSmem*>(smem_raw);
  const int tid = threadIdx.x;
  const int b = blockIdx.x;

  const unsigned m = mask[b] & ~1u;   // class 0 excluded (background)
  if (m == 0u) return;                // uniform: nothing to emit, out already zeroed
  const int cl = __ffs(m) - 1;        // first class >=1 above threshold

  // ---- stage scores for class `cl` into LDS as monotone keys ----
#if HAVE_TDM
  {
    // TDM: one DMA of the strided score column conf[b, 0:P, cl] into LDS.
    // 2D tile: tile_dim0 = 1 elem (4B), tile_dim1 = P rows, row stride = 21 elems.
    const float* gp = conf + ((size_t)b * P_PRI) * C_CLS + cl;
    unsigned long long ga = (unsigned long long)(uintptr_t)gp;
    unsigned ldsoff = (unsigned)(uintptr_t)(l_int*)&S.keys[0];
    u32x4 g0;
    g0.x = 1u;                                      // count=1 (valid user descriptor)
    g0.y = ldsoff;                                  // lds_addr (bytes)
    g0.z = (unsigned)(ga & 0xFFFFFFFFull);          // global_addr[31:0]
    g0.w = (unsigned)((ga >> 32) & 0x01FFFFFFull)   // global_addr[56:32]
           | 0x80000000u;                           // type = 2 ("image")
    i32x8 g1;
    g1[0] = 0x00020000;                             // data_size = 4B (code 2)
    g1[1] = (int)(21u << 16);                       // tensor_dim0 = 21 (bits 79:48 lo)
    g1[2] = (int)(((unsigned)P_PRI & 0xFFFFu) << 16); // tensor_dim1 lo16 (bits 111:80)
    g1[3] = 0x00010000;                             // tile_dim0 = 1 (bits 127:112)
    g1[4] = P_PRI;                                  // tile_dim1 = P, tile_dim2 = 0
    g1[5] = C_CLS;                                  // tensor_dim0_stride = 21 elems
    g1[6] = 0;
    g1[7] = 0;
    i32x4 gz  = {0, 0, 0, 0};                       // groups 2/3 unused (2D tensor)
    i32x8 gz8 = {0, 0, 0, 0, 0, 0, 0, 0};           // extra group (6-arg toolchain form)
    if (tid < 32) {                                 // single wave issues the DMA
      __builtin_amdgcn_tensor_load_to_lds(g0, g1, gz, gz, gz8, 0);
      __builtin_amdgcn_s_wait_tensorcnt(0);
    }
    __syncthreads();
    for (int p = tid; p < P_PRI; p += NT) {
      float s = __uint_as_float(S.keys[p]);
      S.keys[p] = (s > CONF_TH) ? mono(s) : 0u;     // in-place float -> radix key
    }
    __syncthreads();
  }
#elif HAVE_ASYNC_LDS
  for (int base = 0; base < P_PRI; base += CHUNK) {
    const int n = min(CHUNK, P_PRI - base);
    for (int i = tid; i < n; i += NT) {
      const float* gp = conf + ((size_t)b * P_PRI + (size_t)(base + i)) * C_CLS + cl;
      __builtin_amdgcn_global_load_async_to_lds_b32(
          (g_int*)gp, (l_int*)&S.stage[i], 0, 0);
    }
    __builtin_amdgcn_s_wait_asynccnt(0);
    __syncthreads();
    for (int i = tid; i < n; i += NT) {
      float s = S.stage[i];
      S.keys[base + i] = (s > CONF_TH) ? mono(s) : 0u;  // 0 == "-inf" slot
    }
    __syncthreads();
  }
#else
  for (int base = 0; base < P_PRI; base += CHUNK) {
    const int n = min(CHUNK, P_PRI - base);
    for (int i = tid; i < n; i += NT) {
      float s = conf[((size_t)b * P_PRI + (size_t)(base + i)) * C_CLS + cl];
      S.keys[base + i] = (s > CONF_TH) ? mono(s) : 0u;
    }
  }
#endif
  __syncthreads();

  // ---- 4-pass radix select: find key T such that exactly TOPK keys are >= T ----
  if (tid == 0) { S.sPrefix = 0u; S.sNeed = TOPK; }
  __syncthreads();
  for (int pass = 0; pass < 4; ++pass) {
    const int shift = 24 - 8 * pass;
    S.hist[tid] = 0u;  // NT == 256 bins
    __syncthreads();
    unsigned hmask = 0u;
    if (pass) hmask = 0xFFFFFFFFu << (32 - 8 * pass);
    const unsigned pfx = S.sPrefix;
    for (int p = tid; p < P_PRI; p += NT) {
      unsigned k = S.keys[p];
      if ((k & hmask) == (pfx & hmask)) atomicAdd(&S.hist[(k >> shift) & 255u], 1u);
    }
    __syncthreads();
    if (tid == 0) {
      unsigned need = S.sNeed, acc = 0u;
      int chosen = 0;
      for (int d = 255; d >= 0; --d) {
        unsigned h = S.hist[d];
        if (acc + h >= need) { chosen = d; break; }
        acc += h;
      }
      S.sPrefix |= (unsigned)chosen << shift;
      S.sNeed = need - acc;
    }
    __syncthreads();
  }
  const unsigned T = S.sPrefix;
  const unsigned needEq = S.sNeed;  // how many keys == T to take (lowest indices: stable argsort)

  // ---- compact candidates: keys > T (definite), keys == T (ties by index) ----
  if (tid == 0) { S.nGT = 0u; S.nEQ = 0u; }
  __syncthreads();
  for (int p = tid; p < P_PRI; p += NT) {
    unsigned k = S.keys[p];
    if (k > T) {
      unsigned pos = atomicAdd(&S.nGT, 1u);
      if (pos < 256u)
        S.cand[pos] = ((unsigned long long)k << 32) |
                      (unsigned long long)(0xFFFFFFFFu - (unsigned)p);
    } else if (k == T) {
      unsigned pos = atomicAdd(&S.nEQ, 1u);
      if (pos < 512u) S.tie[pos] = (unsigned)p;
    }
  }
  __syncthreads();
  const unsigned nGT = min(S.nGT, (unsigned)TOPK);
  const unsigned nEQst = min(S.nEQ, 512u);
  for (int i = tid; i < 512; i += NT)
    if ((unsigned)i >= nEQst) S.tie[i] = 0xFFFFFFFFu;
  __syncthreads();

  // ascending bitonic sort of tie indices (512 elems, 256 threads)
  for (int ks = 2; ks <= 512; ks <<= 1)
    for (int j = ks >> 1; j > 0; j >>= 1) {
      for (int i = tid; i < 512; i += NT) {
        int ixj = i ^ j;
        if (ixj > i) {
          bool adir = ((i & ks) == 0);
          unsigned a = S.tie[i], c = S.tie[ixj];
          if ((a > c) == adir) { S.tie[i] = c; S.tie[ixj] = a; }
        }
      }
      __syncthreads();
    }

  const unsigned take = min(needEq, nEQst);
  if ((unsigned)tid < take)
    S.cand[nGT + tid] = ((unsigned long long)T << 32) |
                        (unsigned long long)(0xFFFFFFFFu - S.tie[tid]);
  if ((unsigned)tid >= nGT + take) S.cand[tid] = 0ull;
  __syncthreads();

  // descending bitonic sort of 256 packed candidates (key desc, index asc on ties)
  for (int ks = 2; ks <= 256; ks <<= 1)
    for (int j = ks >> 1; j > 0; j >>= 1) {
      int i = tid, ixj = i ^ j;
      if (ixj > i) {
        bool ddir = ((i & ks) == 0);
        unsigned long long a = S.cand[i], c = S.cand[ixj];
        if ((a < c) == ddir) { S.cand[i] = c; S.cand[ixj] = a; }
      }
      __syncthreads();
    }

  // ---- decode boxes for the top-200, SSD center/variance decode ----
  if (tid < TOPK) {
    unsigned long long cd = S.cand[tid];
    unsigned key = (unsigned)(cd >> 32);
    unsigned idx = 0xFFFFFFFFu - (unsigned)(cd & 0xFFFFFFFFull);
    int ok = (key != 0u) && (idx < (unsigned)P_PRI);
    S.validf[tid] = ok;
    S.keepf[tid] = 0;
    float x0 = 0.f, y0 = 0.f, x1 = 0.f, y1 = 0.f, a = 0.f, s = 0.f;
    if (ok) {
      const float* lp = loc + ((size_t)b * P_PRI + idx) * 4;
      const float* pp = prior + (size_t)idx * 4;
      float l0 = lp[0], l1 = lp[1], l2 = lp[2], l3 = lp[3];
      float p0 = pp[0], p1 = pp[1], p2 = pp[2], p3 = pp[3];
      float cx = p0 + l0 * VAR0 * p2;
      float cy = p1 + l1 * VAR0 * p3;
      float w = p2 * __expf(l2 * VAR1);
      float h = p3 * __expf(l3 * VAR1);
      x0 = cx - 0.5f * w; y0 = cy - 0.5f * h;
      x1 = cx + 0.5f * w; y1 = cy + 0.5f * h;
      a = (x1 - x0) * (y1 - y0);
      s = unmono(key);
    }
    S.bx0[tid] = x0; S.by0[tid] = y0; S.bx1[tid] = x1; S.by1[tid] = y1;
    S.ar[tid] = a;   S.sc[tid] = s;
  }
  __syncthreads();

  // ---- sequential greedy NMS: suppress i if IoU>thr with any kept j<i ----
  for (int i = 0; i < TOPK; ++i) {
    if (tid == 0) S.supflag = 0;
    __syncthreads();
    if (tid < i && S.keepf[tid]) {
      float lx = fmaxf(S.bx0[tid], S.bx0[i]);
      float ly = fmaxf(S.by0[tid], S.by0[i]);
      float rx = fminf(S.bx1[tid], S.bx1[i]);
      float ry = fminf(S.by1[tid], S.by1[i]);
      float iw = fmaxf(rx - lx, 0.f), ih = fmaxf(ry - ly, 0.f);
      float inter = iw * ih;
      float iou = inter / (S.ar[tid] + S.ar[i] - inter);
      if (iou > NMS_TH) atomicOr(&S.supflag, 1);
    }
    __syncthreads();
    if (tid == 0) S.keepf[i] = (S.validf[i] && !S.supflag) ? 1 : 0;
    __syncthreads();
  }

  // ---- compact kept rows to the front of out[b, cl] (rest stays zero) ----
  if (tid == 0) {
    int dst = 0;
    float* obase = out + (((size_t)b * C_CLS + cl) * TOPK) * 5;
    for (int k = 0; k < TOPK; ++k) {
      if (S.keepf[k]) {
        float* o = obase + (size_t)dst * 5;
        o[0] = S.sc[k];
        o[1] = S.bx0[k]; o[2] = S.by0[k]; o[3] = S.bx1[k]; o[4] = S.by1[k];
        ++dst;
      }
    }
  }
}

extern "C" void kernel_launch(void* const* d_in, const int* in_sizes, int n_in,
                              void* d_out, int out_size, void* d_ws, size_t ws_size,
                              hipStream_t stream) {
  const float* loc   = (const float*)d_in[0];   // [64, 24564, 4]
  const float* conf  = (const float*)d_in[1];   // [64, 24564, 21]
  const float* prior = (const float*)d_in[2];   // [24564, 4]
  float* out = (float*)d_out;                   // [64, 21, 200, 5]
  unsigned* mask = (unsigned*)d_ws;             // 64 x u32 class bitmasks

  const int initBlocks = (out_size + NT - 1) / NT;
  k_init<<<dim3(initBlocks), dim3(NT), 0, stream>>>(out, out_size, mask);

  const int chunks = (P_PRI + NT - 1) / NT;     // 96
  k_classmask<<<dim3(chunks, B_IMG), dim3(NT), 0, stream>>>(conf, mask);

  k_select<<<dim3(B_IMG), dim3(NT), sizeof(SelSmem), stream>>>(loc, conf, prior, mask, out);
}